// MyGCNLayer_74019466379479
// MI455X (gfx1250) — compile-verified
//
#include <hip/hip_runtime.h>
#include <hip/hip_bf16.h>

#define N_NODES 100000
#define D_IN    256
#define D_OUT   128
#define N_EDGES 1600000

#define LDS_STRIDE 260   // 256 + 4 pad floats -> half-wave rows hit distinct banks

typedef __attribute__((ext_vector_type(2))) float v2f;
typedef __attribute__((ext_vector_type(8))) float v8f;

// ---------------------------------------------------------------------------
// Phase 1: h = (x * dropout_mask) @ W   using V_WMMA_F32_16X16X4_F32
// Grid: 6250 blocks of 256 threads (8 waves). Block b owns rows [16b,16b+16).
// The dropout-applied A tile (16x256) is staged in LDS once per block; each
// wave then computes one 16-col tile of D_OUT=128 from ds_load'ed A operands.
// ---------------------------------------------------------------------------
__global__ __launch_bounds__(256)
void gcn_gemm_wmma_f32(const float* __restrict__ x,
                       const float* __restrict__ mask,
                       const float* __restrict__ W,   // [D_IN][D_OUT] row-major
                       float* __restrict__ h)         // [N_NODES][D_OUT]
{
    __shared__ float sA[16 * LDS_STRIDE];             // 16.25 KB

    const int t       = threadIdx.x;
    const int lane    = t & 31;
    const int wave    = t >> 5;                       // 0..7 -> col tile
    const int rowbase = blockIdx.x * 16;
    const int lh = lane & 15;                         // 0..15
    const int hh = lane >> 4;                         // 0 or 1 (K half)

    // ---- Stage A = x * mask into LDS (1024 float4 chunks, 4 per thread) ----
#pragma unroll
    for (int ch = t; ch < 1024; ch += 256) {
        const int row = ch >> 6;                      // 64 float4 per row
        const int c4  = ch & 63;
        const float4* __restrict__ xr =
            (const float4*)(x    + (size_t)(rowbase + row) * D_IN);
        const float4* __restrict__ mr =
            (const float4*)(mask + (size_t)(rowbase + row) * D_IN);
        const float4 xv = xr[c4];
        const float4 mv = mr[c4];
        float4 av;
        av.x = xv.x * mv.x;  av.y = xv.y * mv.y;
        av.z = xv.z * mv.z;  av.w = xv.w * mv.w;
        // row stride 1040 B and 16 B element offsets are both 16-aligned
        *(float4*)&sA[row * LDS_STRIDE + c4 * 4] = av;  // ds_store_b128
    }
    __syncthreads();

    // ---- WMMA: wave w computes cols [16w, 16w+16) ----
    const int col = wave * 16 + lh;                   // B/N column for this lane
    v8f acc = {};

#pragma unroll 4
    for (int k = 0; k < D_IN; k += 4) {
        const int k0 = k + hh * 2;                    // this lane's K pair
        // A: 16x4 fp32 tile from LDS (8-byte aligned pair -> ds_load_b64)
        const v2f a = *(const v2f*)&sA[lh * LDS_STRIDE + k0];
        // B: 4x16 fp32 tile, rows k0 / k0+1, column `col` (L2/L0-cached W)
        v2f b;
        b.x = W[(size_t)(k0 + 0) * D_OUT + col];
        b.y = W[(size_t)(k0 + 1) * D_OUT + col];
        // 8 args: (neg_a, A, neg_b, B, c_mod, C, reuse_a, reuse_b)
        acc = __builtin_amdgcn_wmma_f32_16x16x4_f32(
                false, a, false, b, (short)0, acc, false, false);
    }

    // C/D layout: VGPR i -> (M = i + 8*hh, N = lh)
#pragma unroll
    for (int i = 0; i < 8; ++i) {
        h[(size_t)(rowbase + hh * 8 + i) * D_OUT + col] = acc[i];
    }
}

// ---------------------------------------------------------------------------
// Phase 2a: zero the accumulation buffer (d_out is poisoned by the harness)
// ---------------------------------------------------------------------------
__global__ __launch_bounds__(256)
void gcn_zero(float4* __restrict__ out, int n4)
{
    int i = blockIdx.x * blockDim.x + threadIdx.x;
    if (i < n4) out[i] = make_float4(0.f, 0.f, 0.f, 0.f);
}

// ---------------------------------------------------------------------------
// Phase 2b: edge scatter-accumulate:  out[row][:] += w_e * h[col][:]
// One wave per edge: 32 lanes x float4 = 128 features. h (51 MB) is
// L2-resident so the random b128 gather runs at L2 bandwidth; accumulation
// uses hardware global_atomic_add_f32 into the L2-resident output.
// ---------------------------------------------------------------------------
__global__ __launch_bounds__(256)
void gcn_edge_scatter(const float4* __restrict__ h4,   // [N_NODES][32] float4
                      const int*    __restrict__ erow,
                      const int*    __restrict__ ecol,
                      const float*  __restrict__ ew,
                      float* __restrict__ out)
{
    const int tid = blockIdx.x * blockDim.x + threadIdx.x;
    const int e  = tid >> 5;            // edge index (one wave per edge)
    const int f4 = tid & 31;            // float4 index within the row
    if (e >= N_EDGES) return;

    const int   r = erow[e];            // wave-uniform
    const int   c = ecol[e];
    const float w = ew[e];

    const float4 hv = h4[(size_t)c * 32 + f4];        // global_load_b128 (L2 hit)
    float* __restrict__ o = out + (size_t)r * D_OUT + f4 * 4;
#if defined(__HIP_DEVICE_COMPILE__)
    unsafeAtomicAdd(o + 0, w * hv.x);                 // global_atomic_add_f32
    unsafeAtomicAdd(o + 1, w * hv.y);
    unsafeAtomicAdd(o + 2, w * hv.z);
    unsafeAtomicAdd(o + 3, w * hv.w);
#else
    atomicAdd(o + 0, w * hv.x);
    atomicAdd(o + 1, w * hv.y);
    atomicAdd(o + 2, w * hv.z);
    atomicAdd(o + 3, w * hv.w);
#endif
}

// ---------------------------------------------------------------------------
// Phase 3: ReLU in place on d_out (vectorized b128)
// ---------------------------------------------------------------------------
__global__ __launch_bounds__(256)
void gcn_relu(float4* __restrict__ out, int n4)
{
    int i = blockIdx.x * blockDim.x + threadIdx.x;
    if (i < n4) {
        float4 v = out[i];
        v.x = fmaxf(v.x, 0.f);  v.y = fmaxf(v.y, 0.f);
        v.z = fmaxf(v.z, 0.f);  v.w = fmaxf(v.w, 0.f);
        out[i] = v;
    }
}

// ---------------------------------------------------------------------------
// Host-side launcher (graph-capture safe: only kernel launches on `stream`)
// ---------------------------------------------------------------------------
extern "C" void kernel_launch(void* const* d_in, const int* in_sizes, int n_in,
                              void* d_out, int out_size, void* d_ws, size_t ws_size,
                              hipStream_t stream)
{
    const float* x    = (const float*)d_in[0];   // [N_NODES, D_IN]
    const float* mask = (const float*)d_in[1];   // [N_NODES, D_IN]
    const float* W    = (const float*)d_in[2];   // [D_IN, D_OUT]
    const int*   er   = (const int*)  d_in[3];   // [N_EDGES]
    const int*   ec   = (const int*)  d_in[4];   // [N_EDGES]
    const float* ewt  = (const float*)d_in[5];   // [N_EDGES]
    float*       out  = (float*)d_out;           // [N_NODES, D_OUT]
    float*       h    = (float*)d_ws;            // scratch: N_NODES*D_OUT f32

    // 1) projection: 6250 row-tiles; LDS-staged A, WMMA f32 16x16x4
    gcn_gemm_wmma_f32<<<N_NODES / 16, 256, 0, stream>>>(x, mask, W, h);

    // 2) zero accumulator, then scatter-add all edge messages
    const int n4 = (N_NODES * D_OUT) / 4;
    gcn_zero<<<(n4 + 255) / 256, 256, 0, stream>>>((float4*)out, n4);

    const long long ethreads = (long long)N_EDGES * 32;
    gcn_edge_scatter<<<(int)((ethreads + 255) / 256), 256, 0, stream>>>(
        (const float4*)h, er, ec, ewt, out);

    // 3) relu
    gcn_relu<<<(n4 + 255) / 256, 256, 0, stream>>>((float4*)out, n4);
}